// DeepGCNCell_25391846654702
// MI455X (gfx1250) — compile-verified
//
#include <hip/hip_runtime.h>

#define N_NODES 50000
#define N_EDGES 800000
#define DIM     256
#define N_RELS  5
#define LDS_STRIDE 260   // 260 % 64 == 4 -> conflict-free column access

typedef __attribute__((ext_vector_type(2))) float v2f;
typedef __attribute__((ext_vector_type(8))) float v8f;

// ---------------------------------------------------------------------------
// Phase 1: per-edge message + scatter-add.  One wave32 per edge; each lane
// handles two float4 chunks (coalesced 512B bursts per wave).  red_sum is
// 51.2 MB and fits in the 192 MB L2, so the f32 atomics resolve at L2.
// ---------------------------------------------------------------------------
__global__ void edge_scatter(const float* __restrict__ h,
                             const int*   __restrict__ src,
                             const int*   __restrict__ dst,
                             const int*   __restrict__ rel,
                             const float* __restrict__ relvec,
                             float*       __restrict__ red_sum,
                             float*       __restrict__ deg) {
  const int wave = threadIdx.x >> 5;
  const int lane = threadIdx.x & 31;
  const int e = blockIdx.x * 8 + wave;            // blockDim == 256 -> 8 waves
  if (e >= N_EDGES) return;

  const int s = src[e];
  const int d = dst[e];
  const int r = rel[e];

  const float4* __restrict__ hs = (const float4*)(h      + (size_t)s * DIM);
  const float4* __restrict__ rv = (const float4*)(relvec + (size_t)r * DIM);
  float* __restrict__ outrow    = red_sum + (size_t)d * DIM;

#pragma unroll
  for (int i = 0; i < 2; ++i) {
    const int c4 = lane + i * 32;                 // float4 index; channel = 4*c4
    float4 a = hs[c4];
    float4 v = rv[c4];
    float m0 = fmaxf(a.x + v.x, 0.0f);
    float m1 = fmaxf(a.y + v.y, 0.0f);
    float m2 = fmaxf(a.z + v.z, 0.0f);
    float m3 = fmaxf(a.w + v.w, 0.0f);
    atomicAdd(outrow + 4 * c4 + 0, m0);
    atomicAdd(outrow + 4 * c4 + 1, m1);
    atomicAdd(outrow + 4 * c4 + 2, m2);
    atomicAdd(outrow + 4 * c4 + 3, m3);
  }
  if (lane == 0) atomicAdd(deg + d, 1.0f);
}

// ---------------------------------------------------------------------------
// Phase 2: h_new = (red_sum / max(deg,1)) @ W^T + b  via V_WMMA_F32_16X16X4_F32
// Block = 512 threads (16 waves) -> one 16x256 output tile per block.
// A tile (16 x 256, degree-scaled) staged in LDS with padded stride.
// Wave w computes output columns [16w, 16w+16).
// ---------------------------------------------------------------------------
__global__ void gemm_wmma(const float* __restrict__ red_sum,
                          const float* __restrict__ deg,
                          const float* __restrict__ W,
                          const float* __restrict__ bias,
                          float*       __restrict__ out) {
  __shared__ float Atile[16 * LDS_STRIDE];

  const int tileM = blockIdx.x * 16;
  const int tid = threadIdx.x;                    // 0..511

  // Stage degree-scaled A tile: coalesced float4 global loads.
  for (int i = tid; i < 16 * (DIM / 4); i += 512) {
    const int m  = i / (DIM / 4);
    const int c4 = i % (DIM / 4);
    const int row = tileM + m;
    const float scale = 1.0f / fmaxf(deg[row], 1.0f);
    float4 a = ((const float4*)(red_sum + (size_t)row * DIM))[c4];
    a.x *= scale; a.y *= scale; a.z *= scale; a.w *= scale;
    *(float4*)&Atile[m * LDS_STRIDE + 4 * c4] = a;
  }
  __syncthreads();

  const int wave  = tid >> 5;
  const int lane  = tid & 31;
  const int tileN = wave * 16;
  const int nm    = lane & 15;                    // row m for A, col n for B
  const int khalf = (lane >> 4) * 2;              // K sub-offset per half-wave

  // B[k][n] = W[(tileN+n)*DIM + k]  (h_new = red @ W^T)
  const float* __restrict__ wrow = W + (size_t)(tileN + nm) * DIM;
  const float* __restrict__ arow = &Atile[nm * LDS_STRIDE];

  v8f acc = {};
#pragma unroll 4
  for (int k0 = 0; k0 < DIM; k0 += 4) {
    const int kk = k0 + khalf;
    v2f a = *(const v2f*)(arow + kk);             // LDS, 8B aligned, no bank conflict
    v2f b = *(const v2f*)(wrow + kk);             // global, L2-resident
    acc = __builtin_amdgcn_wmma_f32_16x16x4_f32(
        /*neg_a=*/false, a, /*neg_b=*/false, b,
        /*c_mod=*/(short)0, acc, /*reuse_a=*/false, /*reuse_b=*/false);
  }

  // C/D layout: lane L, VGPR r -> row tileM + r + (L/16)*8, col tileN + (L%16)
  const float bv = bias[tileN + nm];
  const int rbase = tileM + (lane >> 4) * 8;
#pragma unroll
  for (int r = 0; r < 8; ++r) {
    out[(size_t)(rbase + r) * DIM + tileN + nm] = acc[r] + bv;
  }
}

// ---------------------------------------------------------------------------
extern "C" void kernel_launch(void* const* d_in, const int* in_sizes, int n_in,
                              void* d_out, int out_size, void* d_ws, size_t ws_size,
                              hipStream_t stream) {
  const float* h      = (const float*)d_in[0];
  const int*   src    = (const int*)  d_in[1];
  const int*   dst    = (const int*)  d_in[2];
  const int*   rel    = (const int*)  d_in[3];
  const float* relvec = (const float*)d_in[4];
  const float* W      = (const float*)d_in[5];
  const float* b      = (const float*)d_in[6];
  float* out = (float*)d_out;

  float* red_sum = (float*)d_ws;                         // N_NODES * DIM
  float* deg     = red_sum + (size_t)N_NODES * DIM;      // N_NODES

  hipMemsetAsync(d_ws, 0,
                 ((size_t)N_NODES * DIM + N_NODES) * sizeof(float), stream);

  edge_scatter<<<(N_EDGES + 7) / 8, 256, 0, stream>>>(
      h, src, dst, rel, relvec, red_sum, deg);

  gemm_wmma<<<N_NODES / 16, 512, 0, stream>>>(
      red_sum, deg, W, b, out);
}